// Node_Internal_Dv_Decoder_28217935135446
// MI455X (gfx1250) — compile-verified
//
#include <hip/hip_runtime.h>

// ---------------------------------------------------------------------------
// MI455X / gfx1250 implementation.
//
// Pipeline (all on `stream`, graph-capture safe):
//   1) zero_out            : d_out = 0                        (sums accumulate in-place)
//   2) edge_scatter        : atomicAdd fij/tij -> d_out       (global_atomic_add_f32, no-return)
//   3) mlp3_wmma           : 3 fused 2-layer MLPs via V_WMMA_F32_16X16X4_F32,
//                            W1 staged TRANSPOSED+PADDED in LDS so each WMMA B
//                            operand is one conflict-free ds_load_b64,
//                            heads fused with shfl_xor reduce (wave32)
//   4) combine             : out_dv = m*sum_f + e ; out_dw = i*sum_t   (in-place)
// ---------------------------------------------------------------------------

#define N_NODES 100000
#define N_EDGES 3200000
#define LATENT  128
#define W1T_STRIDE 132   // padded column stride (floats): conflict-free b64 B loads

typedef __attribute__((ext_vector_type(2))) float v2f;
typedef __attribute__((ext_vector_type(8))) float v8f;

// ---------------------------------------------------------------------------
// Kernel 1: zero the output/accumulator buffer (must run every launch).
// ---------------------------------------------------------------------------
__global__ __launch_bounds__(256) void zero_out_kernel(float* __restrict__ p, int n) {
  int i = blockIdx.x * 256 + threadIdx.x;
  if (i < n) p[i] = 0.0f;
}

// ---------------------------------------------------------------------------
// Kernel 2: edge scatter-add.  sum_f/sum_t are [N,3] accumulators (in d_out).
// 3.2M edges * 6 non-returning f32 atomics, all landing in a 2.4MB L2-resident
// region (192MB L2). Streaming reads of fij/tij/receivers are coalesced.
// ---------------------------------------------------------------------------
__global__ __launch_bounds__(256) void edge_scatter_kernel(
    const int*   __restrict__ recv,   // edge_index row 1, [E]
    const float* __restrict__ fij,    // [E,3]
    const float* __restrict__ tij,    // [E,3]
    float* __restrict__ sum_f,        // [N,3]
    float* __restrict__ sum_t) {      // [N,3]
  int e = blockIdx.x * 256 + threadIdx.x;
  if (e >= N_EDGES) return;
  int r = recv[e];
  long eb = 3L * e;
  long rb = 3L * r;
  float f0 = fij[eb + 0], f1 = fij[eb + 1], f2 = fij[eb + 2];
  float t0 = tij[eb + 0], t1 = tij[eb + 1], t2 = tij[eb + 2];
  atomicAdd(&sum_f[rb + 0], f0);
  atomicAdd(&sum_f[rb + 1], f1);
  atomicAdd(&sum_f[rb + 2], f2);
  atomicAdd(&sum_t[rb + 0], t0);
  atomicAdd(&sum_t[rb + 1], t1);
  atomicAdd(&sum_t[rb + 2], t2);
}

// ---------------------------------------------------------------------------
// Kernel 3: three fused 2-layer MLPs with fp32 WMMA.
//
// Block = 256 threads = 8 waves; wave w owns 16 nodes.  A tile (16x128 f32)
// lives in 32 v2f registers per lane (ISA A-layout: lane%16 = M,
// VGPR pair kk = K{4kk+2h, 4kk+2h+1} with h = lane/16).
//
// W1 is staged TRANSPOSED into LDS with padded stride 132 floats:
//   w1t[n*132 + k] = W1[k*128 + n]
// so a lane's B operand (same column n, K pair {4kk+2h, 4kk+2h+1}) is one
// aligned ds_load_b64; bank index = (4*li + 2h + r0) mod 64 covers all 64
// banks exactly once per wave -> conflict-free, and the pair lands in an
// adjacent even-aligned VGPR pair (no repacking movs for WMMA SRC1).
//
// After K accumulation, bias+ReLU+W2-dot are fused in the C layout
// (VGPR r: M = r + 8h, N = lane%16) and reduced across the 16-lane half
// with __shfl_xor (wave32).  Result scalars -> d_ws as [3][N].
// ---------------------------------------------------------------------------
__global__ __launch_bounds__(256) void mlp3_wmma_kernel(
    const float* __restrict__ X,
    const float* __restrict__ mW1, const float* __restrict__ mb1,
    const float* __restrict__ mW2, const float* __restrict__ mb2,
    const float* __restrict__ iW1, const float* __restrict__ ib1,
    const float* __restrict__ iW2, const float* __restrict__ ib2,
    const float* __restrict__ eW1, const float* __restrict__ eb1,
    const float* __restrict__ eW2, const float* __restrict__ eb2,
    float* __restrict__ scalars /* [3][N_NODES] */) {

  __shared__ float w1t[LATENT * W1T_STRIDE];  // 67.6 KB of the 320 KB WGP LDS

  const int tid  = threadIdx.x;
  const int lane = tid & 31;
  const int wave = tid >> 5;
  const int half = lane >> 4;   // 0: lanes 0-15, 1: lanes 16-31
  const int li   = lane & 15;
  const int nodeBase = blockIdx.x * 128 + wave * 16;

  // ---- load A tile: 16 nodes x 128 features, resident for all 3 MLPs ----
  v2f a[32];
  {
    int node = nodeBase + li;
    if (node >= N_NODES) node = N_NODES - 1;          // clamp: keep EXEC all-1s
    const v2f* row = (const v2f*)(X + (long)node * LATENT);  // 512B-aligned row
#pragma unroll
    for (int kk = 0; kk < 32; ++kk)
      a[kk] = row[2 * kk + half];                     // K = {4kk+2h, 4kk+2h+1}
  }

  const float* W1p[3] = {mW1, iW1, eW1};
  const float* b1p[3] = {mb1, ib1, eb1};
  const float* W2p[3] = {mW2, iW2, eW2};
  const float* b2p[3] = {mb2, ib2, eb2};

  for (int mlp = 0; mlp < 3; ++mlp) {
    __syncthreads();  // protect w1t readers from previous iteration
    // ---- cooperative transpose-stage of W1 into LDS ----
    // Coalesced float4 reads along n; scattered b32 LDS writes down columns.
    {
      const float4* src = (const float4*)W1p[mlp];
#pragma unroll
      for (int i = 0; i < 16; ++i) {
        int flat = i * 256 + tid;          // 4096 float4 chunks
        int k    = flat >> 5;              // row (k) 0..127
        int nb   = (flat & 31) * 4;        // first of 4 columns
        float4 v = src[flat];
        w1t[(nb + 0) * W1T_STRIDE + k] = v.x;
        w1t[(nb + 1) * W1T_STRIDE + k] = v.y;
        w1t[(nb + 2) * W1T_STRIDE + k] = v.z;
        w1t[(nb + 3) * W1T_STRIDE + k] = v.w;
      }
    }
    // Prefetch next MLP's W1 into cache while we compute (global_prefetch_b8).
    if (mlp < 2) {
      const char* nxt = (const char*)W1p[mlp + 1];
      __builtin_prefetch(nxt + (size_t)tid * 128, 0, 1);
      __builtin_prefetch(nxt + 32768 + (size_t)tid * 128, 0, 1);
    }
    __syncthreads();

    const float* b1 = b1p[mlp];
    const float* w2 = W2p[mlp];
    float ps[8] = {0.f, 0.f, 0.f, 0.f, 0.f, 0.f, 0.f, 0.f};

#pragma unroll
    for (int t = 0; t < 8; ++t) {          // 8 N-tiles of 16 hidden units
      const int n = t * 16 + li;           // this lane's hidden column
      const v2f* bcol = (const v2f*)(w1t + n * W1T_STRIDE);  // 8B-aligned (132 even)
      v8f c = {0.f, 0.f, 0.f, 0.f, 0.f, 0.f, 0.f, 0.f};
#pragma unroll
      for (int kk = 0; kk < 32; ++kk) {    // K = 128 in steps of 4
        v2f b = bcol[2 * kk + half];       // K = {4kk+2h, 4kk+2h+1}: one ds_load_b64
        c = __builtin_amdgcn_wmma_f32_16x16x4_f32(
                false, a[kk], false, b, (short)0, c, false, false);
      }
      // fused head: h = relu(c + b1[n]); ps += h * W2[n]
      const float bias = b1[n];
      const float w2v  = w2[n];            // W2 is [128,1]
#pragma unroll
      for (int r = 0; r < 8; ++r) {
        float h = c[r] + bias;
        h = h > 0.f ? h : 0.f;
        ps[r] += h * w2v;
      }
    }

    // ---- reduce over the 16 lanes of each half (N dimension) ----
#pragma unroll
    for (int r = 0; r < 8; ++r) {
      float v = ps[r];
      v += __shfl_xor(v, 1, 32);
      v += __shfl_xor(v, 2, 32);
      v += __shfl_xor(v, 4, 32);
      v += __shfl_xor(v, 8, 32);
      ps[r] = v;
    }
    const float b2v = b2p[mlp][0];
#pragma unroll
    for (int r = 0; r < 8; ++r) {
      if (li == r) {                       // one lane per half writes M = r + 8h
        int node = nodeBase + r + 8 * half;
        if (node < N_NODES)
          scalars[(long)mlp * N_NODES + node] = ps[r] + b2v;
      }
    }
  }
}

// ---------------------------------------------------------------------------
// Kernel 4: in-place combine.
//   out_dv[n,c] = m_inv[n]*sum_f[n,c] + dv_ext[n];  out_dw[n,c] = i_inv[n]*sum_t[n,c]
// ---------------------------------------------------------------------------
__global__ __launch_bounds__(256) void combine_kernel(
    const float* __restrict__ scalars,  // [3][N]
    float* __restrict__ out) {          // [2][N,3], sums already in place
  int idx = blockIdx.x * 256 + threadIdx.x;
  const int total = N_NODES * 3;
  if (idx >= total) return;
  int n = idx / 3;
  float m_inv  = scalars[n];
  float i_inv  = scalars[N_NODES + n];
  float dv_ext = scalars[2 * N_NODES + n];
  out[idx]         = m_inv * out[idx] + dv_ext;
  out[total + idx] = i_inv * out[total + idx];
}

// ---------------------------------------------------------------------------
extern "C" void kernel_launch(void* const* d_in, const int* in_sizes, int n_in,
                              void* d_out, int out_size, void* d_ws, size_t ws_size,
                              hipStream_t stream) {
  const int*   edge_index = (const int*)  d_in[0];   // [2, E] int32
  const float* X          = (const float*)d_in[1];   // [N, 128]
  const float* fij        = (const float*)d_in[2];   // [E, 3]
  const float* tij        = (const float*)d_in[3];   // [E, 3]
  const float* mW1 = (const float*)d_in[4];
  const float* mb1 = (const float*)d_in[5];
  const float* mW2 = (const float*)d_in[6];
  const float* mb2 = (const float*)d_in[7];
  const float* iW1 = (const float*)d_in[8];
  const float* ib1 = (const float*)d_in[9];
  const float* iW2 = (const float*)d_in[10];
  const float* ib2 = (const float*)d_in[11];
  const float* eW1 = (const float*)d_in[12];
  const float* eb1 = (const float*)d_in[13];
  const float* eW2 = (const float*)d_in[14];
  const float* eb2 = (const float*)d_in[15];

  float* out     = (float*)d_out;          // [2][N,3] = 600000 floats
  float* scalars = (float*)d_ws;           // [3][N]   = 1.2 MB scratch
  float* sum_f   = out;                    // accumulate segment sums in-place
  float* sum_t   = out + (long)N_NODES * 3;

  const int* recv = edge_index + N_EDGES;  // row 1 = receivers

  const int total_out = N_NODES * 3 * 2;
  zero_out_kernel<<<(total_out + 255) / 256, 256, 0, stream>>>(out, total_out);

  edge_scatter_kernel<<<(N_EDGES + 255) / 256, 256, 0, stream>>>(
      recv, fij, tij, sum_f, sum_t);

  const int mlp_blocks = (N_NODES + 127) / 128;  // 128 nodes per 256-thread block
  mlp3_wmma_kernel<<<mlp_blocks, 256, 0, stream>>>(
      X, mW1, mb1, mW2, mb2, iW1, ib1, iW2, ib2, eW1, eb1, eW2, eb2, scalars);

  combine_kernel<<<(N_NODES * 3 + 255) / 256, 256, 0, stream>>>(scalars, out);
}